// GNN_65987877536243
// MI455X (gfx1250) — compile-verified
//
#include <hip/hip_runtime.h>

// ---------------------------------------------------------------------------
// 3-layer GCN forward for MI455X (gfx1250, wave32).
//   layer: out = relu( D^-1/2 (A+I) D^-1/2 (x W) + b )
// Restructured:
//   hs   = dinv ⊙ (x @ W)                       (WMMA f32 16x16x4 GEMM)
//   acc[c] += ew_e * hs[row_e]   for all edges  (native f32 L2 atomics)
//   out[c] = relu( dinv[c]*(acc[c] + hs[c]) + b )   (self-loop folded in)
// ---------------------------------------------------------------------------

typedef __attribute__((ext_vector_type(2))) float v2f;
typedef __attribute__((ext_vector_type(8))) float v8f;

// Guaranteed-native CDNA5 f32 atomic add (non-returning, device scope).
// Non-returning form is tracked on STOREcnt; S_ENDPGM's implicit wait-idle
// guarantees completion before the next kernel on the stream.
__device__ __forceinline__ void atomic_add_f32_dev(float* p, float v) {
  asm volatile("global_atomic_add_f32 %0, %1, off scope:SCOPE_DEV"
               :
               : "v"(p), "v"(v)
               : "memory");
}

// ---- degree / normalization -----------------------------------------------

__global__ __launch_bounds__(256) void gcn_deg_init(float* __restrict__ deg, int n) {
  int i = blockIdx.x * blockDim.x + threadIdx.x;
  if (i < n) deg[i] = 1.0f;  // self-loop weight
}

__global__ __launch_bounds__(256) void gcn_deg_accum(const long long* __restrict__ col,
                                                     const float* __restrict__ ew,
                                                     float* __restrict__ deg, int e) {
  int i = blockIdx.x * blockDim.x + threadIdx.x;
  if (i < e) atomic_add_f32_dev(&deg[(int)col[i]], ew[i]);
}

__global__ __launch_bounds__(256) void gcn_make_dinv(float* __restrict__ deg, int n) {
  int i = blockIdx.x * blockDim.x + threadIdx.x;
  if (i < n) {
    float d = deg[i];
    deg[i] = (d > 0.0f) ? rsqrtf(d) : 0.0f;
  }
}

// ---- GEMM: out[N,16] = dinv ⊙ (A[N,K] @ W[K,16]) via V_WMMA_F32_16X16X4_F32 --
// One wave handles one 16-node tile. EXEC is all-ones for every WMMA
// (tile-level early-out is wave-uniform; tail rows are clamped on load and
// predicated only on the stores after the WMMA loop).

template <int K>
__global__ __launch_bounds__(256) void gcn_gemm_scale_wmma(
    const float* __restrict__ A, const float* __restrict__ W,
    const float* __restrict__ dinv, float* __restrict__ out, int n) {
  const int lane   = threadIdx.x & 31;
  const int wave   = threadIdx.x >> 5;
  const int tile   = blockIdx.x * 8 + wave;
  const int ntiles = (n + 15) >> 4;
  if (tile >= ntiles) return;  // wave-uniform: EXEC stays full for WMMA

  const int half = lane >> 4;   // 0: lanes 0-15, 1: lanes 16-31
  const int l16  = lane & 15;

  int arow = tile * 16 + l16;
  if (arow >= n) arow = n - 1;  // clamp tail loads; discarded on store

  v8f c = {};
#pragma unroll
  for (int k0 = 0; k0 < K; k0 += 4) {
    // A 16x4 (MxK): lane half selects K pair {0,1} or {2,3}
    v2f a = *(const v2f*)(A + (size_t)arow * K + (k0 + half * 2));
    // B 4x16 (KxN): row striped across lanes; lane half selects K pair
    v2f b;
    b.x = W[(k0 + half * 2 + 0) * 16 + l16];
    b.y = W[(k0 + half * 2 + 1) * 16 + l16];
    c = __builtin_amdgcn_wmma_f32_16x16x4_f32(
        /*neg_a=*/false, a, /*neg_b=*/false, b,
        /*c_mod=*/(short)0, c, /*reuse_a=*/false, /*reuse_b=*/false);
  }

  // C 16x16: VGPR r -> row (r + 8*half), lane%16 -> column
#pragma unroll
  for (int r = 0; r < 8; ++r) {
    int node = tile * 16 + r + half * 8;
    if (node < n) out[(size_t)node * 16 + l16] = c[r] * dinv[node];
  }
}

// ---- zero accumulator ------------------------------------------------------

__global__ __launch_bounds__(256) void gcn_zero4(float4* __restrict__ p, int n4) {
  int i = blockIdx.x * blockDim.x + threadIdx.x;
  if (i < n4) p[i] = make_float4(0.f, 0.f, 0.f, 0.f);
}

// ---- edge scatter: acc[col] += ew * hs[row], 4 lanes per edge --------------

__global__ __launch_bounds__(256) void gcn_edge_scatter(
    const long long* __restrict__ row, const long long* __restrict__ col,
    const float* __restrict__ ew, const float* __restrict__ hs,
    float* __restrict__ acc, int e) {
  int tid = blockIdx.x * blockDim.x + threadIdx.x;
  int ei  = tid >> 2;
  if (ei >= e) return;
  int c0 = (tid & 3) * 4;
  int r  = (int)row[ei];
  int cl = (int)col[ei];
  float w = ew[ei];
  float4 v = *(const float4*)(hs + (size_t)r * 16 + c0);
  float* dst = acc + (size_t)cl * 16 + c0;
  atomic_add_f32_dev(dst + 0, w * v.x);
  atomic_add_f32_dev(dst + 1, w * v.y);
  atomic_add_f32_dev(dst + 2, w * v.z);
  atomic_add_f32_dev(dst + 3, w * v.w);
}

// ---- finalize layer (hidden): acc := relu(dinv*(acc + hs) + b) -------------

__global__ __launch_bounds__(256) void gcn_finalize_relu(
    float* __restrict__ acc, const float* __restrict__ hs,
    const float* __restrict__ dinv, const float* __restrict__ b, int n) {
  int tid = blockIdx.x * blockDim.x + threadIdx.x;
  if (tid < n * 16) {
    int node = tid >> 4;
    int ch   = tid & 15;
    float v  = dinv[node] * (acc[tid] + hs[tid]) + b[ch];
    acc[tid] = fmaxf(v, 0.0f);
  }
}

// ---- finalize layer 2 fused with output projection (16 -> 1) ---------------

__global__ __launch_bounds__(256) void gcn_output_proj(
    const float* __restrict__ acc, const float* __restrict__ hs,
    const float* __restrict__ dinv, const float* __restrict__ b2,
    const float* __restrict__ Wout, const float* __restrict__ bout,
    float* __restrict__ out, int n) {
  int node = blockIdx.x * blockDim.x + threadIdx.x;
  if (node >= n) return;
  float di  = dinv[node];
  float sum = bout[0];
#pragma unroll
  for (int c = 0; c < 16; ++c) {
    size_t idx = (size_t)node * 16 + c;
    float h2 = fmaxf(di * (acc[idx] + hs[idx]) + b2[c], 0.0f);
    sum += h2 * Wout[c];
  }
  out[node] = sum;
}

// ---------------------------------------------------------------------------

extern "C" void kernel_launch(void* const* d_in, const int* in_sizes, int n_in,
                              void* d_out, int out_size, void* d_ws, size_t ws_size,
                              hipStream_t stream) {
  // inputs (setup_inputs order):
  // 0:x[N,128] 1:edge_index[2,E] i64 2:edge_attr[E] 3:y 4:batch
  // 5:W1[128,16] 6:b1[16] 7:W2[16,16] 8:b2[16] 9:Wout[16,1] 10:bout[1]
  const float*     x    = (const float*)d_in[0];
  const long long* ei   = (const long long*)d_in[1];
  const float*     ew   = (const float*)d_in[2];
  const float*     W1   = (const float*)d_in[5];
  const float*     b1   = (const float*)d_in[6];
  const float*     W2   = (const float*)d_in[7];
  const float*     b2   = (const float*)d_in[8];
  const float*     Wout = (const float*)d_in[9];
  const float*     bout = (const float*)d_in[10];
  float*           out  = (float*)d_out;

  const int N = in_sizes[0] / 128;
  const int E = in_sizes[1] / 2;
  const long long* row = ei;
  const long long* col = ei + E;

  // workspace: dinv [N, padded] | hs [N*16] | acc [N*16]
  float* dinv = (float*)d_ws;
  float* hs   = dinv + (((size_t)N + 255) & ~(size_t)255);
  float* acc  = hs + (size_t)N * 16;

  const int ntiles    = (N + 15) / 16;
  const int gemm_grid = (ntiles + 7) / 8;
  const int nch       = N * 16;
  const int edge_grid = (E * 4 + 255) / 256;

  // normalization
  gcn_deg_init<<<(N + 255) / 256, 256, 0, stream>>>(dinv, N);
  gcn_deg_accum<<<(E + 255) / 256, 256, 0, stream>>>(col, ew, dinv, E);
  gcn_make_dinv<<<(N + 255) / 256, 256, 0, stream>>>(dinv, N);

  // layer 1: hs = dinv ⊙ (x@W1); acc = scatter; acc := relu(dinv*(acc+hs)+b1)
  gcn_gemm_scale_wmma<128><<<gemm_grid, 256, 0, stream>>>(x, W1, dinv, hs, N);
  gcn_zero4<<<(nch / 4 + 255) / 256, 256, 0, stream>>>((float4*)acc, nch / 4);
  gcn_edge_scatter<<<edge_grid, 256, 0, stream>>>(row, col, ew, hs, acc, E);
  gcn_finalize_relu<<<(nch + 255) / 256, 256, 0, stream>>>(acc, hs, dinv, b1, N);

  // layer 2: hs = dinv ⊙ (h1@W2); acc = scatter; fused finalize + projection
  gcn_gemm_scale_wmma<16><<<gemm_grid, 256, 0, stream>>>(acc, W2, dinv, hs, N);
  gcn_zero4<<<(nch / 4 + 255) / 256, 256, 0, stream>>>((float4*)acc, nch / 4);
  gcn_edge_scatter<<<edge_grid, 256, 0, stream>>>(row, col, ew, hs, acc, E);
  gcn_output_proj<<<(N + 255) / 256, 256, 0, stream>>>(acc, hs, dinv, b2, Wout,
                                                       bout, out, N);
}